// SampleCrossAttention_6244882448560
// MI455X (gfx1250) — compile-verified
//
#include <hip/hip_runtime.h>
#include <hip/hip_bf16.h>
#include <stdint.h>

#define FEAT   256
#define HEADS  8
#define HD     32
#define POS    128
#define NEG    128
#define HW     16384
#define BB     8
#define NSLOT  512
#define SS     256         // POS + NEG
#define LN_EPS 1e-5f

typedef float v2f __attribute__((ext_vector_type(2)));
typedef float v8f __attribute__((ext_vector_type(8)));

// ---------------------------------------------------------------------------
// Kernel 1: per-slot bitonic sort of the 16384-entry curiosity map in LDS.
// Key = (monotone float bits << 32) | (16383 - idx)  -> descending sort with
// stable tie-break on ascending index (matches jnp.argsort(-flat)).
// Emits rowmap[s*NSLOT + n] = feat_idx[s][n]*B + batch_idx[n] for s in [0,256):
//   s <  128 : rank s               (pos_idx, top-128)
//   s >= 128 : rank 128+127*(s-128) (neg_idx, every 127th of the pool)
// 128 KB dynamic LDS per block (CDNA5 WGP has 320 KB -> 2 blocks/WGP).
// ---------------------------------------------------------------------------
__global__ __launch_bounds__(1024)
void sort_select_kernel(const float* __restrict__ curio,
                        const int* __restrict__ batch_idx,
                        int* __restrict__ rowmap)
{
    extern __shared__ unsigned long long keys[];
    const int n   = blockIdx.x;
    const int tid = threadIdx.x;
    const float* src = curio + (size_t)n * HW;

    for (int i = tid; i < HW; i += 1024) {
        unsigned int fb = __float_as_uint(src[i]);
        unsigned int u  = fb ^ ((fb & 0x80000000u) ? 0xFFFFFFFFu : 0x80000000u);
        keys[i] = ((unsigned long long)u << 32) |
                  (unsigned long long)(16383u - (unsigned)i);
    }
    __syncthreads();

    for (int k = 2; k <= HW; k <<= 1) {
        for (int j = k >> 1; j > 0; j >>= 1) {
            for (int t = tid; t < (HW / 2); t += 1024) {
                int low = t & (j - 1);
                int i   = ((t & ~(j - 1)) << 1) | low;
                int p   = i | j;
                unsigned long long A = keys[i];
                unsigned long long Bv = keys[p];
                bool up = (i & k) == 0;         // descending blocks
                bool sw = up ? (A < Bv) : (A > Bv);
                if (sw) { keys[i] = Bv; keys[p] = A; }
            }
            __syncthreads();
        }
    }

    const int bidx = batch_idx[n];
    for (int s = tid; s < SS; s += 1024) {
        int rank = (s < POS) ? s : (POS + 127 * (s - POS));
        unsigned long long key = keys[rank];
        int pos = 16383 - (int)(key & 0xFFFFFFFFull);
        rowmap[s * NSLOT + n] = pos * BB + bidx;
    }
}

// ---------------------------------------------------------------------------
// Kernel 2: generic (gather-)GEMM
//   Y[m, g] = (sum_f X[src(m), f] (+ X2[src(m), f]) * W[g, f] + b[g]) * scale
// using V_WMMA_F32_16X16X4_F32 (f32 matrix path -> bit-matches reference
// precision while exercising the CDNA5 matrix pipe). One wave owns a
// 16(M) x 128(N) tile = 8 f32 accumulators; A fragment reused 8x; every A/B
// element loaded exactly once per wave. grid = (M/64, 2), block = 128 (4 waves).
// HAS_X2 specialized at compile time so the pos_encodings add is branch-free
// (no per-iteration saveexec in the hot loop).
// ---------------------------------------------------------------------------
template <bool HAS_X2, bool HAS_MAP>
__global__ __launch_bounds__(128)
void gemm_xwT_kernel(const float* __restrict__ X,
                     const float* __restrict__ X2,     // second addend (pos_encodings)
                     const int* __restrict__ rowmap,   // row gather map
                     const float* __restrict__ W,      // [256 x 256], row-major (g, f)
                     const float* __restrict__ Bias,   // [256]
                     float* __restrict__ Y,            // [M x 256]
                     float scale)
{
    const int wave = threadIdx.x >> 5;
    const int lane = threadIdx.x & 31;
    const int m0   = (blockIdx.x * 4 + wave) * 16;
    const int n0   = blockIdx.y * 128;

    const int mr   = m0 + (lane & 15);
    const int src  = HAS_MAP ? rowmap[mr] : mr;
    const float* xr  = X + (size_t)src * FEAT;
    const float* x2r = HAS_X2 ? (X2 + (size_t)src * FEAT) : nullptr;
    const int khi  = (lane >> 4) << 1;        // 0 or 2

    v8f acc[8] = {};

    for (int kk = 0; kk < FEAT; kk += 4) {
        // Speculative prefetch of upcoming A data (dropped silently past row end).
        __builtin_prefetch((const void*)(xr + kk + 16), 0, 3); // global_prefetch_b8
        v2f a;
        a.x = xr[kk + khi];
        a.y = xr[kk + khi + 1];
        if (HAS_X2) { a.x += x2r[kk + khi]; a.y += x2r[kk + khi + 1]; }
#pragma unroll
        for (int j = 0; j < 8; ++j) {
            const int col = n0 + j * 16 + (lane & 15);
            v2f b;
            b.x = W[(size_t)col * FEAT + kk + khi];
            b.y = W[(size_t)col * FEAT + kk + khi + 1];
            acc[j] = __builtin_amdgcn_wmma_f32_16x16x4_f32(
                false, a, false, b, (short)0, acc[j], false, false);
        }
    }

    const int rbase = m0 + ((lane >> 4) << 3);
#pragma unroll
    for (int j = 0; j < 8; ++j) {
        const int col = n0 + j * 16 + (lane & 15);
        const float bv = Bias[col];
#pragma unroll
        for (int r = 0; r < 8; ++r) {
            Y[(size_t)(rbase + r) * FEAT + col] = (acc[j][r] + bv) * scale;
        }
    }
}

// ---------------------------------------------------------------------------
// Kernel 3: affinities[s, n, h] = sum_d q[n, h*32+d] * kg[(s*512+n), h*32+d]
// One block per slot n; thread t = key index s; q row cached in LDS.
// ---------------------------------------------------------------------------
__global__ __launch_bounds__(256)
void affinity_kernel(const float* __restrict__ q,
                     const float* __restrict__ kg,
                     float* __restrict__ aff)      // [SS, NSLOT, HEADS]
{
    __shared__ float qs[FEAT];
    const int n = blockIdx.x;
    const int t = threadIdx.x;
    qs[t] = q[(size_t)n * FEAT + t];
    __syncthreads();

    const int s = t;
    const float* kr = kg + ((size_t)s * NSLOT + n) * FEAT;
#pragma unroll
    for (int h = 0; h < HEADS; ++h) {
        float sum = 0.0f;
#pragma unroll
        for (int d = 0; d < HD; ++d)
            sum += qs[h * HD + d] * kr[h * HD + d];
        aff[(size_t)s * (NSLOT * HEADS) + (size_t)n * HEADS + h] = sum;
    }
}

// ---------------------------------------------------------------------------
// Kernel 4: softmax over p in [0,128) per (n, h), then
// attn_out[n, c] = sum_p attn[p, h(c)] * vg[(p*512+n), c]
// One block per slot n, 256 threads.
// ---------------------------------------------------------------------------
__global__ __launch_bounds__(256)
void softmax_attnout_kernel(const float* __restrict__ aff,
                            const float* __restrict__ vg,
                            float* __restrict__ attn_out) // [NSLOT, FEAT]
{
    __shared__ float at[HEADS][POS];  // 4 KB
    const int n = blockIdx.x;
    const int t = threadIdx.x;

    for (int i = t; i < HEADS * POS; i += 256) {
        int p = i & (POS - 1);
        int h = i >> 7;
        at[h][p] = aff[(size_t)p * (NSLOT * HEADS) + (size_t)n * HEADS + h];
    }
    __syncthreads();

    if (t < HEADS) {
        float m = -3.0e38f;
        for (int p = 0; p < POS; ++p) m = fmaxf(m, at[t][p]);
        float sum = 0.0f;
        for (int p = 0; p < POS; ++p) {
            float e = expf(at[t][p] - m);
            at[t][p] = e;
            sum += e;
        }
        float inv = 1.0f / sum;
        for (int p = 0; p < POS; ++p) at[t][p] *= inv;
    }
    __syncthreads();

    const int c = t;
    const int h = c >> 5;
    float s = 0.0f;
    for (int p = 0; p < POS; ++p)
        s += at[h][p] * vg[((size_t)p * NSLOT + n) * FEAT + c];
    attn_out[(size_t)n * FEAT + c] = s;
}

// ---------------------------------------------------------------------------
// Kernel 5: x = slots + proj; LayerNorm over FEAT; write new_slots.
// ---------------------------------------------------------------------------
__global__ __launch_bounds__(256)
void residual_ln_kernel(const float* __restrict__ slots,
                        const float* __restrict__ proj,
                        const float* __restrict__ g,
                        const float* __restrict__ b,
                        float* __restrict__ out)
{
    __shared__ float red[256];
    const int n = blockIdx.x;
    const int t = threadIdx.x;
    const float x = slots[(size_t)n * FEAT + t] + proj[(size_t)n * FEAT + t];

    red[t] = x;
    __syncthreads();
    for (int off = 128; off > 0; off >>= 1) {
        if (t < off) red[t] += red[t + off];
        __syncthreads();
    }
    const float mu = red[0] * (1.0f / FEAT);
    __syncthreads();

    const float d = x - mu;
    red[t] = d * d;
    __syncthreads();
    for (int off = 128; off > 0; off >>= 1) {
        if (t < off) red[t] += red[t + off];
        __syncthreads();
    }
    const float var = red[0] * (1.0f / FEAT);

    out[(size_t)n * FEAT + t] = d * rsqrtf(var + LN_EPS) * g[t] + b[t];
}

// ---------------------------------------------------------------------------
extern "C" void kernel_launch(void* const* d_in, const int* in_sizes, int n_in,
                              void* d_out, int out_size, void* d_ws, size_t ws_size,
                              hipStream_t stream)
{
    (void)in_sizes; (void)n_in; (void)out_size; (void)ws_size;

    const float* slots     = (const float*)d_in[0];
    const float* features  = (const float*)d_in[1];
    const float* pos_enc   = (const float*)d_in[2];
    const float* curio     = (const float*)d_in[3];
    const int*   batch_idx = (const int*)  d_in[4];
    const float* in_w      = (const float*)d_in[5];   // [3*256, 256]
    const float* in_b      = (const float*)d_in[6];   // [3*256]
    const float* out_w     = (const float*)d_in[7];
    const float* out_b     = (const float*)d_in[8];
    const float* ln_g      = (const float*)d_in[9];
    const float* ln_b      = (const float*)d_in[10];

    // workspace layout
    char* ws = (char*)d_ws;
    int*   rowmap = (int*)ws;   ws += (size_t)SS * NSLOT * sizeof(int);          // 512 KB
    float* q      = (float*)ws; ws += (size_t)NSLOT * FEAT * sizeof(float);      // 512 KB
    float* attn_o = (float*)ws; ws += (size_t)NSLOT * FEAT * sizeof(float);      // 512 KB
    float* proj   = (float*)ws; ws += (size_t)NSLOT * FEAT * sizeof(float);      // 512 KB
    float* kg     = (float*)ws; ws += (size_t)SS * NSLOT * FEAT * sizeof(float); // 128 MB
    float* vg     = (float*)ws;                                                  // 64 MB

    float* new_slots = (float*)d_out;                        // [512*256]
    float* aff       = (float*)d_out + (size_t)NSLOT * FEAT; // [256*512*8]

    const float scaling = 0.17677669529663687f; // HD^-0.5 = 1/sqrt(32)

    // 1) argsort + rank selection -> gather rowmap
    hipLaunchKernelGGL(sort_select_kernel, dim3(NSLOT), dim3(1024),
                       (size_t)HW * sizeof(unsigned long long), stream,
                       curio, batch_idx, rowmap);

    // 2) q = (slots @ qw^T + qb) * scaling        [512 x 256]
    gemm_xwT_kernel<false, false><<<dim3(NSLOT / 64, 2), 128, 0, stream>>>(
        slots, nullptr, nullptr, in_w, in_b, q, scaling);

    // 3) kg = (features + pos_enc)[gather] @ kw^T + kb    [131072 x 256]
    gemm_xwT_kernel<true, true><<<dim3((SS * NSLOT) / 64, 2), 128, 0, stream>>>(
        features, pos_enc, rowmap, in_w + FEAT * FEAT, in_b + FEAT, kg, 1.0f);

    // 4) vg = features[gather(:POS)] @ vw^T + vb          [65536 x 256]
    gemm_xwT_kernel<false, true><<<dim3((POS * NSLOT) / 64, 2), 128, 0, stream>>>(
        features, nullptr, rowmap, in_w + 2 * FEAT * FEAT, in_b + 2 * FEAT, vg, 1.0f);

    // 5) affinities (second output, written directly to d_out)
    affinity_kernel<<<dim3(NSLOT), 256, 0, stream>>>(q, kg, aff);

    // 6) softmax over POS + weighted V sum
    softmax_attnout_kernel<<<dim3(NSLOT), 256, 0, stream>>>(aff, vg, attn_o);

    // 7) out projection
    gemm_xwT_kernel<false, false><<<dim3(NSLOT / 64, 2), 128, 0, stream>>>(
        attn_o, nullptr, nullptr, out_w, out_b, proj, 1.0f);

    // 8) residual + LayerNorm -> new_slots (first output)
    residual_ln_kernel<<<dim3(NSLOT), 256, 0, stream>>>(
        slots, proj, ln_g, ln_b, new_slots);
}